// PolyPoolerTextLenSensitive_43250320671416
// MI455X (gfx1250) — compile-verified
//
#include <hip/hip_runtime.h>
#include <cmath>

typedef __attribute__((ext_vector_type(2))) float v2f;
typedef __attribute__((ext_vector_type(8))) float v8f;

#define N_POLY 14
#define C_CH   256
// output region sizes (elements): N*C*8*32 and N*C*8*64
#define REGION0_ELEMS ((size_t)256 * 256 * 8 * 32)   // 16,777,216

__global__ __launch_bounds__(256)
void poly_pool_kernel(const float* __restrict__ f0,
                      const float* __restrict__ f1,
                      const float* __restrict__ f2,
                      const float* __restrict__ f3,
                      const float* __restrict__ polys,
                      const int*   __restrict__ img_ids,
                      const int*   __restrict__ lens,
                      float*       __restrict__ out)
{
    // B-matrix lookup table for the WMMA: s_ctrl[N*8 + k] = B[k, N]
    //   N=0: top x, N=1: top y, N=2: bot x, N=3: bot y (normalized); N>=4 or k==7 -> 0
    __shared__ float s_ctrl[16 * 8];
    __shared__ int   s_meta[4];              // lvl, pid, img
    __shared__ float s_bnd[64 * 4];          // per x: tx, ty, bx, by
    __shared__ int   s_off[512 * 4];         // per sample point: 4 texel offsets
    __shared__ float s_wgt[512 * 4];         // per sample point: 4 bilinear weights (0 if OOB)

    const int n   = blockIdx.x;
    const int tid = threadIdx.x;

    // ---------- Phase 0: fill WMMA B-table directly from polys; per-instance routing ----------
    if (tid < 128) {
        const int N = tid >> 3;              // column 0..15
        const int k = tid & 7;               // row 0..7
        float v = 0.0f;
        if (N < 4 && k < 7) {
            // top control k = raw point k; bottom control k = raw point 13-k (reversed)
            const int raw = (N < 2) ? k : (13 - k);
            const float px = polys[(n * N_POLY + raw) * 2 + 0];
            const float py = polys[(n * N_POLY + raw) * 2 + 1];
            v = (N & 1) ? (py * (1.0f / 800.0f))     // wh = (336/0.25, 200/0.25)
                        : (px * (1.0f / 1344.0f));
        }
        s_ctrl[tid] = v;
    }
    if (tid == 0) {
        float mnx = 1e30f, mxx = -1e30f, mny = 1e30f, mxy = -1e30f;
        for (int i = 0; i < N_POLY; ++i) {
            float px = polys[(n * N_POLY + i) * 2 + 0];
            float py = polys[(n * N_POLY + i) * 2 + 1];
            mnx = fminf(mnx, px); mxx = fmaxf(mxx, px);
            mny = fminf(mny, py); mxy = fmaxf(mxy, py);
        }
        float s  = sqrtf((mxx - mnx) * (mxy - mny));
        float lf = floorf(4.0f + log2f(s * (1.0f / 224.0f) + 1e-6f));
        lf = fminf(fmaxf(lf, 2.0f), 5.0f);
        s_meta[0] = (int)lf - 2;                 // feature level 0..3
        s_meta[1] = (lens[n] > 8) ? 1 : 0;       // pooler id
        s_meta[2] = img_ids[n];
    }
    __syncthreads();

    const int lvl = s_meta[0];
    const int pid = s_meta[1];
    const int img = s_meta[2];
    const int wp  = (pid == 0) ? 32 : 64;
    const int P   = 8 * wp;                      // samples per instance (256 or 512)
    const int HL  = 200 >> lvl;
    const int WL  = 336 >> lvl;

    // ---------- Phase 1: boundary interpolation as GEMM via V_WMMA_F32_16X16X4_F32 ----------
    // D(16 x-positions, 4 cols {tx,ty,bx,by}) = A(16x8 interp weights) x B(8x4 control pts),
    // split into two K=4 WMMAs accumulated through C.
    const int wave   = tid >> 5;
    const int lane   = tid & 31;
    const int nTiles = wp >> 4;                  // 2 or 4 (wave-uniform branch, EXEC all-1)
    if (wave < nTiles) {
        const int   M      = lane & 15;          // row of A / col of B per ISA layout
        const int   hi     = (lane >= 16) ? 1 : 0;
        const int   xq     = wave * 16 + M;
        const float uscale = (pid == 0) ? (6.0f / 32.0f) : (6.0f / 64.0f);
        float u  = ((float)xq + 0.5f) * uscale;
        int   i0 = (int)floorf(u);
        i0 = (i0 < 0) ? 0 : ((i0 > 5) ? 5 : i0);
        float fr = u - (float)i0;

        v8f acc = {};
        #pragma unroll
        for (int chunk = 0; chunk < 2; ++chunk) {
            const int kb  = chunk * 4 + hi * 2;  // this lane's two K slots (A layout 16x4 f32)
            const int ka0 = kb, ka1 = kb + 1;
            v2f A, Bv;
            A.x = (ka0 == i0) ? (1.0f - fr) : ((ka0 == i0 + 1) ? fr : 0.0f);
            A.y = (ka1 == i0) ? (1.0f - fr) : ((ka1 == i0 + 1) ? fr : 0.0f);
            // B 4x16: lane%16 = N column; half-wave+VGPR selects K row — flat LDS table lookup
            Bv.x = s_ctrl[(M << 3) | ka0];
            Bv.y = s_ctrl[(M << 3) | ka1];
            acc = __builtin_amdgcn_wmma_f32_16x16x4_f32(
                      false, A, false, Bv, (short)0, acc, false, false);
        }
        // D layout: lane L holds cols N=L%16, rows M=j+(L<16?0:8) in VGPR j
        if (M < 4) {
            #pragma unroll
            for (int j = 0; j < 8; ++j) {
                int Mm = j + hi * 8;
                s_bnd[(wave * 16 + Mm) * 4 + M] = acc[j];
            }
        }
    }
    __syncthreads();

    // ---------- Phase 2: per-sample offsets & bilinear weights into LDS ----------
    const int wsh = (pid == 0) ? 5 : 6;
    for (int p = tid; p < P; p += 256) {
        int y = p >> wsh, x = p & (wp - 1);
        float tx = s_bnd[x * 4 + 0], ty = s_bnd[x * 4 + 1];
        float bx = s_bnd[x * 4 + 2], by = s_bnd[x * 4 + 3];
        float v  = ((float)y + 0.5f) * (1.0f / 8.0f);
        float gx = tx + (bx - tx) * v;
        float gy = ty + (by - ty) * v;
        float X  = gx * (float)WL - 0.5f;
        float Y  = gy * (float)HL - 0.5f;
        float xf = floorf(X), yf = floorf(Y);
        float wx = X - xf,   wy = Y - yf;
        int x0 = (int)xf, y0 = (int)yf;
        #pragma unroll
        for (int k = 0; k < 4; ++k) {
            int xi = x0 + (k & 1), yi = y0 + (k >> 1);
            int xc = xi < 0 ? 0 : (xi > WL - 1 ? WL - 1 : xi);
            int yc = yi < 0 ? 0 : (yi > HL - 1 ? HL - 1 : yi);
            bool in = (xi >= 0) && (xi < WL) && (yi >= 0) && (yi < HL);
            float wk = ((k & 1) ? wx : 1.0f - wx) * ((k >> 1) ? wy : 1.0f - wy);
            s_off[p * 4 + k] = yc * WL + xc;
            s_wgt[p * 4 + k] = in ? wk : 0.0f;
        }
    }
    __syncthreads();

    // ---------- Phase 3: stream channels, gather 4 texels, blend, coalesced store ----------
    const float* fsel = (lvl == 0) ? f0 : (lvl == 1) ? f1 : (lvl == 2) ? f2 : f3;
    const int    HW   = HL * WL;
    const float* base = fsel + (size_t)img * (size_t)C_CH * (size_t)HW;
    const size_t actBase = ((pid == 0) ? (size_t)0 : REGION0_ELEMS)
                         + (size_t)n * (size_t)C_CH * (size_t)P;

    const bool two = (P == 512);
    int   oa0 = s_off[tid * 4 + 0], oa1 = s_off[tid * 4 + 1];
    int   oa2 = s_off[tid * 4 + 2], oa3 = s_off[tid * 4 + 3];
    float wa0 = s_wgt[tid * 4 + 0], wa1 = s_wgt[tid * 4 + 1];
    float wa2 = s_wgt[tid * 4 + 2], wa3 = s_wgt[tid * 4 + 3];
    int   ob0 = oa0, ob1 = oa1, ob2 = oa2, ob3 = oa3;
    float wb0 = 0.f, wb1 = 0.f, wb2 = 0.f, wb3 = 0.f;
    if (two) {
        int pb = tid + 256;
        ob0 = s_off[pb * 4 + 0]; ob1 = s_off[pb * 4 + 1];
        ob2 = s_off[pb * 4 + 2]; ob3 = s_off[pb * 4 + 3];
        wb0 = s_wgt[pb * 4 + 0]; wb1 = s_wgt[pb * 4 + 1];
        wb2 = s_wgt[pb * 4 + 2]; wb3 = s_wgt[pb * 4 + 3];
    }

    for (int c = 0; c < C_CH; ++c) {
        const float* ch = base + (size_t)c * HW;
        if (c + 2 < C_CH) {
            __builtin_prefetch(ch + 2 * HW + oa0, 0, 1);   // global_prefetch_b8
        }
        float a = wa0 * ch[oa0] + wa1 * ch[oa1] + wa2 * ch[oa2] + wa3 * ch[oa3];
        out[actBase + (size_t)c * P + tid] = a;
        if (two) {
            float b = wb0 * ch[ob0] + wb1 * ch[ob1] + wb2 * ch[ob2] + wb3 * ch[ob3];
            out[actBase + (size_t)c * P + tid + 256] = b;
        }
    }

    // ---------- Phase 4: zero this instance's slice in the inactive pooler region ----------
    {
        const int    Po = (pid == 0) ? 512 : 256;
        const size_t ib = ((pid == 0) ? REGION0_ELEMS : (size_t)0)
                        + (size_t)n * (size_t)C_CH * (size_t)Po;
        float4* o4 = (float4*)(out + ib);
        const int t4 = (C_CH * Po) >> 2;
        float4 z = make_float4(0.f, 0.f, 0.f, 0.f);
        for (int i = tid; i < t4; i += 256) o4[i] = z;
    }
}

extern "C" void kernel_launch(void* const* d_in, const int* in_sizes, int n_in,
                              void* d_out, int out_size, void* d_ws, size_t ws_size,
                              hipStream_t stream) {
    const float* f0      = (const float*)d_in[0];
    const float* f1      = (const float*)d_in[1];
    const float* f2      = (const float*)d_in[2];
    const float* f3      = (const float*)d_in[3];
    const float* polys   = (const float*)d_in[4];
    const int*   img_ids = (const int*)d_in[5];
    const int*   lens    = (const int*)d_in[6];
    float*       out     = (float*)d_out;

    const int N = in_sizes[5];   // 256 instances; one workgroup (8 wave32) per instance
    poly_pool_kernel<<<N, 256, 0, stream>>>(f0, f1, f2, f3, polys, img_ids, lens, out);
}